// LightningFastPPO_65970697666687
// MI455X (gfx1250) — compile-verified
//
#include <hip/hip_runtime.h>
#include <hip/hip_bf16.h>

// ---------------------------------------------------------------------------
// LightningFastPPO (2-layer GCN + skip + heads) for MI455X / gfx1250.
// Sparse aggregation dominates traffic but stays L2-resident (H/AGG = 25.6MB
// each vs 192MB L2). Dense GEMMs use V_WMMA_F32_16X16X4_F32 with W staged in
// LDS (bank-conflict-free) and A slabs preloaded to registers, so the WMMA
// chain never waits on global memory.
// ---------------------------------------------------------------------------

typedef __attribute__((ext_vector_type(2))) float v2f;
typedef __attribute__((ext_vector_type(8))) float v8f;

#define IN_CH   64
#define HIDDEN  128
#define ACTIONS 64

// ---------------------------------------------------------------------------
// C[N, OUT] = A[N, K] @ W[OUT, K]^T  via V_WMMA_F32_16X16X4_F32 (wave32).
// Block = 128 threads (4 waves); each wave owns a 16-row x OUT stripe.
// W is staged in LDS in 64-column slabs (pad to 68 floats/row -> lane(mr,half)
// hits bank 4*mr+2*half: all 32 lanes on disjoint bank pairs, conflict-free).
// A fragment (16x4 f32): lane L holds A[m0 + L%16][kb + 2*(L/16) + j], j=0,1.
// B fragment (4x16):     lane L holds W[t*16 + L%16][kb + 2*(L/16) + j].
// D layout: lane L, vgpr v -> C[m0 + v + 8*(L/16)][t*16 + L%16].
// ---------------------------------------------------------------------------
template <int K, int OUT>
__global__ void gemm_nt_wmma(const float* __restrict__ A,
                             const float* __restrict__ W,
                             float* __restrict__ C, int N) {
    constexpr int KS  = 64;          // K-slab staged in LDS
    constexpr int LDW = KS + 4;      // padded row stride (floats)
    constexpr int CT  = OUT / 16;    // column tiles
    __shared__ float Wlds[OUT][LDW];

    const int lane    = threadIdx.x & 31;
    const int wave    = threadIdx.x >> 5;
    const int rowTile = blockIdx.x * (blockDim.x >> 5) + wave;
    const int m0      = rowTile * 16;
    const bool active = (m0 < N);    // wave-uniform; keep barriers block-wide

    const int half = lane >> 4;      // 0 -> K offset +0, 1 -> +2
    const int mr   = lane & 15;      // row (A) / col (B,D) within tile

    v8f acc[CT] = {};

    int arow = m0 + mr;
    if (arow >= N) arow = N - 1;     // clamp: garbage rows never stored
    const float* Arow = A + (size_t)arow * K;

    for (int ks = 0; ks < K; ks += KS) {
        // ---- cooperative, coalesced W[:, ks:ks+KS] -> LDS ----
        __syncthreads();             // protect previous slab's readers
        for (int i = threadIdx.x * 4; i < OUT * KS; i += blockDim.x * 4) {
            int o = i / KS, c = i % KS;
            *(float4*)&Wlds[o][c] = *(const float4*)&W[(size_t)o * K + ks + c];
        }

        // ---- preload A slab fragments (one clause, overlaps LDS fill) ----
        v2f afrag[KS / 4];
        if (active) {
#pragma unroll
            for (int kk = 0; kk < KS / 4; ++kk)
                afrag[kk] = *(const v2f*)(Arow + ks + kk * 4 + half * 2);
        }
        __syncthreads();

        // ---- WMMA over the slab: B from LDS only ----
        if (active) {
#pragma unroll
            for (int kk = 0; kk < KS / 4; ++kk) {
                v2f b[CT];
#pragma unroll
                for (int t = 0; t < CT; ++t)
                    b[t] = *(const v2f*)&Wlds[t * 16 + mr][kk * 4 + half * 2];
#pragma unroll
                for (int t = 0; t < CT; ++t)
                    acc[t] = __builtin_amdgcn_wmma_f32_16x16x4_f32(
                        false, afrag[kk], false, b[t], (short)0, acc[t], false, false);
            }
        }
    }

    if (active) {
#pragma unroll
        for (int t = 0; t < CT; ++t) {
#pragma unroll
            for (int v = 0; v < 8; ++v) {
                int m = m0 + v + half * 8;
                if (m < N) C[(size_t)m * OUT + t * 16 + mr] = acc[t][v];
            }
        }
    }
}

// ---------------------------------------------------------------------------
// Degree count over targets (self-loop +1 folded into dinv kernel).
// ---------------------------------------------------------------------------
__global__ void deg_kernel(const int* __restrict__ tgt, int* __restrict__ deg, int E) {
    int e = blockIdx.x * blockDim.x + threadIdx.x;
    if (e < E) atomicAdd(&deg[tgt[e]], 1);
}

__global__ void dinv_kernel(const int* __restrict__ deg, float* __restrict__ dinv, int N) {
    int i = blockIdx.x * blockDim.x + threadIdx.x;
    if (i < N) dinv[i] = rsqrtf((float)(deg[i] + 1));   // +1 = self loop; always > 0
}

// ---------------------------------------------------------------------------
// Edge scatter: one wave per edge. Lane l covers channels 4l..4l+3 (float4
// gather = one 512B coalesced row per wave), then 4 f32 atomics to AGG[tgt].
// Both H and AGG are L2-resident (25.6MB each vs 192MB L2).
// ---------------------------------------------------------------------------
__global__ void scatter_edges(const float* __restrict__ H,
                              const int* __restrict__ src,
                              const int* __restrict__ tgt,
                              const float* __restrict__ dinv,
                              float* __restrict__ AGG, int E) {
    int gw   = (int)((blockIdx.x * blockDim.x + threadIdx.x) >> 5);
    int lane = threadIdx.x & 31;
    if (gw >= E) return;
    int s = src[gw];
    int t = tgt[gw];
    float norm = dinv[s] * dinv[t];
    float4 v = ((const float4*)(H + (size_t)s * HIDDEN))[lane];
    float* arow = AGG + (size_t)t * HIDDEN + lane * 4;
    atomicAdd(arow + 0, v.x * norm);
    atomicAdd(arow + 1, v.y * norm);
    atomicAdd(arow + 2, v.z * norm);
    atomicAdd(arow + 3, v.w * norm);
}

// ---------------------------------------------------------------------------
// Fused epilogue: AGG = relu(AGG + b + dinv^2*H (self loop) + SKIP [+ bskip]).
// ---------------------------------------------------------------------------
__global__ void combine_relu(float* __restrict__ AGG,
                             const float* __restrict__ H,
                             const float* __restrict__ SKIP,
                             const float* __restrict__ b,
                             const float* __restrict__ bskip,
                             const float* __restrict__ dinv,
                             long long total) {
    long long i = (long long)blockIdx.x * blockDim.x + threadIdx.x;
    if (i >= total) return;
    int node = (int)(i >> 7);
    int c    = (int)(i & (HIDDEN - 1));
    float d  = dinv[node];
    float v  = AGG[i] + b[c] + d * d * H[i] + SKIP[i];
    if (bskip) v += bskip[c];
    AGG[i] = v > 0.f ? v : 0.f;
}

// ---------------------------------------------------------------------------
// Mean pool, stage 1: block sums 64 rows per channel, one atomic per channel.
// ---------------------------------------------------------------------------
__global__ void pool_reduce(const float* __restrict__ H, float* __restrict__ pooled, int N) {
    int c  = threadIdx.x;                 // blockDim.x == HIDDEN
    int r0 = blockIdx.x * 64;
    int r1 = r0 + 64; if (r1 > N) r1 = N;
    float s = 0.f;
    for (int r = r0; r < r1; ++r) s += H[(size_t)r * HIDDEN + c];
    atomicAdd(&pooled[c], s);
}

// ---------------------------------------------------------------------------
// Heads: out[0..63] = pooled/N @ Wa^T + ba ; out[64] = pooled/N @ Wc^T + bc.
// ---------------------------------------------------------------------------
__global__ void head_kernel(const float* __restrict__ pooled,
                            const float* __restrict__ Wa, const float* __restrict__ ba,
                            const float* __restrict__ Wc, const float* __restrict__ bc,
                            float* __restrict__ out, float invN) {
    int t = threadIdx.x;
    if (t < ACTIONS) {
        float s = 0.f;
        for (int h = 0; h < HIDDEN; ++h) s += pooled[h] * invN * Wa[t * HIDDEN + h];
        out[t] = s + ba[t];
    } else if (t == ACTIONS) {
        float s = 0.f;
        for (int h = 0; h < HIDDEN; ++h) s += pooled[h] * invN * Wc[h];
        out[ACTIONS] = s + bc[0];
    }
}

// ---------------------------------------------------------------------------
extern "C" void kernel_launch(void* const* d_in, const int* in_sizes, int n_in,
                              void* d_out, int out_size, void* d_ws, size_t ws_size,
                              hipStream_t stream) {
    const float* x     = (const float*)d_in[0];
    const int*   ei    = (const int*)d_in[1];      // [2, E] flat: src then tgt
    const float* W1    = (const float*)d_in[2];
    const float* b1    = (const float*)d_in[3];
    const float* W2    = (const float*)d_in[4];
    const float* b2    = (const float*)d_in[5];
    const float* Wskip = (const float*)d_in[6];
    const float* bskip = (const float*)d_in[7];
    const float* Wa    = (const float*)d_in[8];
    const float* ba    = (const float*)d_in[9];
    const float* Wc    = (const float*)d_in[10];
    const float* bc    = (const float*)d_in[11];
    float* out = (float*)d_out;

    const int N = in_sizes[0] / IN_CH;
    const int E = in_sizes[1] / 2;
    const int* src = ei;
    const int* tgt = ei + E;

    // Workspace carve-up (all f32 unless noted)
    const size_t NH = (size_t)N * HIDDEN;
    float* bufH   = (float*)d_ws;        // GEMM out: H1, later H2
    float* bufS   = bufH + NH;           // SKIP, later AGG2 / final h2
    float* bufA   = bufS + NH;           // AGG1, later h (layer-2 input/identity)
    float* dinv   = bufA + NH;           // [N]
    int*   deg    = (int*)(dinv + N);    // [N]
    float* pooled = (float*)(deg + N);   // [HIDDEN]

    const long long total = (long long)NH;
    const int cBlocks   = (int)((total + 255) / 256);
    const int rowTiles  = (N + 15) / 16;
    const int gemmGrid  = (rowTiles + 3) / 4;          // 4 waves / block
    const int edgeGrid  = (E + 7) / 8;                 // 8 waves (edges) / block
    const int poolGrid  = (N + 63) / 64;

    // ---- degree / normalization ----
    hipMemsetAsync(deg, 0, (size_t)N * sizeof(int), stream);
    deg_kernel<<<(E + 255) / 256, 256, 0, stream>>>(tgt, deg, E);
    dinv_kernel<<<(N + 255) / 256, 256, 0, stream>>>(deg, dinv, N);

    // ---- layer 1 dense parts (WMMA) ----
    gemm_nt_wmma<IN_CH, HIDDEN><<<gemmGrid, 128, 0, stream>>>(x, W1, bufH, N);
    gemm_nt_wmma<IN_CH, HIDDEN><<<gemmGrid, 128, 0, stream>>>(x, Wskip, bufS, N);

    // ---- layer 1 aggregation ----
    hipMemsetAsync(bufA, 0, NH * sizeof(float), stream);
    scatter_edges<<<edgeGrid, 256, 0, stream>>>(bufH, src, tgt, dinv, bufA, E);
    combine_relu<<<cBlocks, 256, 0, stream>>>(bufA, bufH, bufS, b1, bskip, dinv, total);

    // ---- layer 2 ----
    gemm_nt_wmma<HIDDEN, HIDDEN><<<gemmGrid, 128, 0, stream>>>(bufA, W2, bufH, N);
    hipMemsetAsync(bufS, 0, NH * sizeof(float), stream);
    scatter_edges<<<edgeGrid, 256, 0, stream>>>(bufH, src, tgt, dinv, bufS, E);
    combine_relu<<<cBlocks, 256, 0, stream>>>(bufS, bufH, bufA, b2, nullptr, dinv, total);

    // ---- pool + heads ----
    hipMemsetAsync(pooled, 0, HIDDEN * sizeof(float), stream);
    pool_reduce<<<poolGrid, HIDDEN, 0, stream>>>(bufS, pooled, N);
    head_kernel<<<1, 128, 0, stream>>>(pooled, Wa, ba, Wc, bc, out, 1.0f / (float)N);
}